// LabelSmoothingCTCLoss_32435593019989
// MI455X (gfx1250) — compile-verified
//
#include <hip/hip_runtime.h>
#include <cstdint>
#include <cstddef>

// Problem constants (from the reference)
#define T_ 1000
#define B_ 64
#define C_ 512
#define S_ 100
#define L_ 201            // 2*S+1 extended states
#define D_ 8              // async pipeline depth (power of two)
#define NEGF (-1e30f)

// ---------------------------------------------------------------------------
// CDNA5 async global->LDS gather (ASYNCcnt path). Each lane loads 4 bytes
// from its own 64-bit global address into its own LDS slot (wave-relative
// LDS byte address in a VGPR). Tracked by ASYNCcnt; in-order completion.
// ---------------------------------------------------------------------------
__device__ __forceinline__ void async_gather_b32(const float* gptr, unsigned lds_byte) {
    asm volatile("global_load_async_to_lds_b32 %0, %1, off"
                 :
                 : "v"(lds_byte), "v"((unsigned long long)(uintptr_t)gptr)
                 : "memory");
}
#define ASYNC_WAIT_7() asm volatile("s_wait_asynccnt 0x7" ::: "memory")
#define ASYNC_WAIT_0() asm volatile("s_wait_asynccnt 0x0" ::: "memory")

__device__ __forceinline__ unsigned lds_addr_of(const void* p) {
    // Generic pointer to __shared__ : low 32 bits are the wave-relative LDS
    // byte address (flat-LDS path adds LDS_BASE in hardware).
    return (unsigned)(uintptr_t)p;
}

// log(exp(a)+exp(b)) with -1e30 acting as -inf (matches jnp.logaddexp usage here)
__device__ __forceinline__ float lse2(float a, float b) {
    float m = fmaxf(a, b);
    float d = fminf(a, b) - m;            // <= 0
    return m + __logf(1.0f + __expf(d));  // arg in (1,2]: well conditioned
}

// ---------------------------------------------------------------------------
// Kernel 1: entropy partial sum.  ws_ent accumulates sum_{t,b,c} exp(lp)*lp.
// Pure streaming pass over 131 MB -> HBM-bound (~5.6 us roofline at 23.3 TB/s).
// ---------------------------------------------------------------------------
__global__ __launch_bounds__(256) void ctc_entropy_kernel(
    const float* __restrict__ lp, float* __restrict__ ws_ent, long long n4) {
    long long i = (long long)blockIdx.x * blockDim.x + threadIdx.x;
    const long long stride = (long long)gridDim.x * blockDim.x;
    const float4* p4 = (const float4*)lp;
    float acc = 0.0f;
    for (; i < n4; i += stride) {
        float4 v = p4[i];
        acc += v.x * __expf(v.x);
        acc += v.y * __expf(v.y);
        acc += v.z * __expf(v.z);
        acc += v.w * __expf(v.w);
    }
    __shared__ float red[256];
    red[threadIdx.x] = acc;
    __syncthreads();
    for (int s = 128; s > 0; s >>= 1) {
        if ((int)threadIdx.x < s) red[threadIdx.x] += red[threadIdx.x + s];
        __syncthreads();
    }
    if (threadIdx.x == 0) atomicAdd(ws_ent, red[0]);
}

// ---------------------------------------------------------------------------
// Kernel 2: CTC alpha recursion. One block (8 wave32) per batch element.
// Thread l <-> extended state l (l<201). lp gathers for step t+D are issued
// asynchronously into LDS slot (t%D) while step t computes -> the 1000-step
// serial chain never stalls on global latency (D=8 loads in flight per wave).
// Steady-state loop body: s_wait_asynccnt 7 -> barrier -> ds_load -> 2x lse
// -> ds_store -> async issue (pointer bump only, no 64-bit mul).
// ws_nll accumulates sum_b nll_b / target_len_b.
// ---------------------------------------------------------------------------
__global__ __launch_bounds__(256) void ctc_alpha_kernel(
    const float* __restrict__ lp, const int* __restrict__ targets,
    const int* __restrict__ input_lengths, const int* __restrict__ target_lengths,
    float* __restrict__ ws_nll) {
    const int b = blockIdx.x;
    const int l = threadIdx.x;

    __shared__ float lpbuf[D_][256];   // async-gather pipeline, slot = t % D
    __shared__ float albuf[2][256];    // alpha double buffer

    // Per-thread extended-sequence class and skip-transition flag.
    int  cls  = 0;       // blank for even l and for padding threads l>=L
    bool skip = false;
    if (l < L_ && (l & 1)) {
        const int s = (l - 1) >> 1;
        cls = targets[b * S_ + s];
        if (l >= 3) {
            const int prevc = targets[b * S_ + s - 1];
            skip = (cls != 0) && (cls != prevc);
        }
    }

    const int myIn = input_lengths[b];
    const int myTl = target_lengths[b];

    // Per-thread global gather base: lp[t, b, cls], stride B*C floats per t.
    const float*   rowbase = lp + (size_t)b * C_ + cls;
    const unsigned ldsbase = lds_addr_of(&lpbuf[0][l]);   // slot stride = 1024 B

    // ---- pipeline prologue: issue t = 0..D-1 ----
    #pragma unroll
    for (int tp = 0; tp < D_; ++tp) {
        async_gather_b32(rowbase + (size_t)tp * (B_ * C_),
                         ldsbase + ((unsigned)tp << 10));
    }

    // ---- t = 0: init alpha ----
    ASYNC_WAIT_7();                         // load t=0 landed (in-order)
    const float lp0 = lpbuf[0][l];
    albuf[0][l] = (l < 2) ? lp0 : NEGF;     // alpha0[0]=lp(blank), alpha0[1]=lp(tgt0)
    albuf[1][l] = NEGF;                     // padding slots l>=L stay NEG forever
    // slot 0 consumed by this lane -> refill with t = D
    async_gather_b32(rowbase + (size_t)D_ * (B_ * C_), ldsbase);

    int p = 0;
    // Running issue pointer for tn = t + D (first issue inside loop is tn = 1+D).
    const float* gq = rowbase + (size_t)(1 + D_) * (B_ * C_);

    // ---- steady state: t = 1 .. T-D-1 (issues tn = D+1 .. T-1) ----
    int t = 1;
    for (; t < T_ - D_; ++t) {
        ASYNC_WAIT_7();                     // loads 0..t complete (in-order)
        __syncthreads();                    // alpha(t-1) visible block-wide
        const float lpv = lpbuf[t & (D_ - 1)][l];
        if (l < L_) {
            const float a0 = albuf[p][l];
            const float a1 = (l >= 1) ? albuf[p][l - 1] : NEGF;
            const float a2 = skip ? albuf[p][l - 2] : NEGF;
            float anew = lse2(lse2(a0, a1), a2) + lpv;
            anew = (t < myIn) ? anew : a0;  // inactive timesteps keep alpha
            albuf[p ^ 1][l] = anew;
        }
        // refill the slot just consumed with t + D
        async_gather_b32(gq, ldsbase + ((unsigned)(t & (D_ - 1)) << 10));
        gq += B_ * C_;
        p ^= 1;
    }
    // ---- drain: last D iterations, nothing left to issue ----
    for (; t < T_; ++t) {
        ASYNC_WAIT_0();
        __syncthreads();
        const float lpv = lpbuf[t & (D_ - 1)][l];
        if (l < L_) {
            const float a0 = albuf[p][l];
            const float a1 = (l >= 1) ? albuf[p][l - 1] : NEGF;
            const float a2 = skip ? albuf[p][l - 2] : NEGF;
            float anew = lse2(lse2(a0, a1), a2) + lpv;
            anew = (t < myIn) ? anew : a0;
            albuf[p ^ 1][l] = anew;
        }
        p ^= 1;
    }
    __syncthreads();

    if (l == 0) {
        const int last = 2 * myTl;          // tl in [50,100] -> last in [100,200]
        const float ll = lse2(albuf[p][last], albuf[p][last - 1]);
        float nll = -ll;
        if (nll > 1e29f) nll = 0.0f;
        atomicAdd(ws_nll, nll / (float)myTl);
    }
}

// ---------------------------------------------------------------------------
// Kernel 3: combine.  out = (1-S)*mean(nll/tl) - S*entropy
//   entropy = -ent_sum/(T*B)  =>  out = 0.9*nll_sum/B + 0.1*ent_sum/(T*B)
// ---------------------------------------------------------------------------
__global__ void ctc_finalize_kernel(const float* __restrict__ ws, float* __restrict__ out) {
    out[0] = 0.9f * (ws[0] / (float)B_) + 0.1f * (ws[1] / ((float)T_ * (float)B_));
}

extern "C" void kernel_launch(void* const* d_in, const int* in_sizes, int n_in,
                              void* d_out, int out_size, void* d_ws, size_t ws_size,
                              hipStream_t stream) {
    const float* lp       = (const float*)d_in[0];   // (T,B,C) f32
    const int*   targets  = (const int*)d_in[1];     // (B,S) i32
    const int*   in_len   = (const int*)d_in[2];     // (B,) i32
    const int*   tgt_len  = (const int*)d_in[3];     // (B,) i32
    float*       out      = (float*)d_out;
    float*       ws       = (float*)d_ws;            // ws[0]=nll_sum, ws[1]=ent_sum

    hipMemsetAsync(ws, 0, 2 * sizeof(float), stream);

    const long long n4 = (long long)T_ * B_ * C_ / 4;
    ctc_entropy_kernel<<<2048, 256, 0, stream>>>(lp, ws + 1, n4);
    ctc_alpha_kernel<<<B_, 256, 0, stream>>>(lp, targets, in_len, tgt_len, ws);
    ctc_finalize_kernel<<<1, 1, 0, stream>>>(ws, out);
}